// GATLayer_64682207478126
// MI455X (gfx1250) — compile-verified
//
#include <hip/hip_runtime.h>

#define DIM   128
#define NH    8
#define HD    16
#define NB    4
#define NN    20000
#define NE    100000
#define LDSS  132   // padded LDS row stride (floats): 132 mod 64 = 4 -> conflict-free A-frag reads

typedef __attribute__((ext_vector_type(2))) float v2f;
typedef __attribute__((ext_vector_type(8))) float v8f;

// ---------- WMMA fragment helpers (V_WMMA_F32_16X16X4_F32 layouts, ISA 7.12.2) ----------
// A 16x4 f32: VGPR0 = K=k0 (lanes 0-15) / K=k0+2 (lanes 16-31); VGPR1 = k0+1 / k0+3
__device__ __forceinline__ v2f ldsA(const float* s, int lane, int k0) {
    int r  = lane & 15;
    int kk = k0 + ((lane >> 4) << 1);
    v2f a;
    a.x = s[r * LDSS + kk];
    a.y = s[r * LDSS + kk + 1];
    return a;
}
// B 4x16 f32: VGPR0 = row k0 (lanes 0-15, N=0..15) / row k0+2 (lanes 16-31); VGPR1 = k0+1 / k0+3
__device__ __forceinline__ v2f glbB(const float* W, int ldw, int col0, int lane, int k0) {
    int c  = col0 + (lane & 15);
    int kk = k0 + ((lane >> 4) << 1);
    v2f b;
    b.x = W[kk * ldw + c];
    b.y = W[(kk + 1) * ldw + c];
    return b;
}

// ---------- ordered-float <-> uint encoding for atomic float max ----------
__device__ __forceinline__ unsigned int fflip(float f) {
    unsigned int u = __float_as_uint(f);
    return (u & 0x80000000u) ? ~u : (u | 0x80000000u);
}
__device__ __forceinline__ float funflip(unsigned int u) {
    unsigned int b = (u & 0x80000000u) ? (u ^ 0x80000000u) : ~u;
    return __uint_as_float(b);
}

// ---------- fold Wq/Wk with attn vectors: weff (128 x 16), beff (16) ----------
// col c<8 : weff[:,c] = Wq[:, c*16:(c+1)*16] @ attn[c, :16]      (aq path)
// col c>=8: weff[:,c] = Wk[:, h*16:(h+1)*16] @ attn[h, 16:32], h=c-8 (ak path)
__global__ void prep_weff(const float* __restrict__ Wq, const float* __restrict__ bq,
                          const float* __restrict__ Wk, const float* __restrict__ bk,
                          const float* __restrict__ attn,
                          float* __restrict__ weff, float* __restrict__ beff) {
    int t = threadIdx.x;
    for (int i = t; i < DIM * 16; i += 256) {
        int d = i >> 4, c = i & 15;
        int h = c & 7;
        const float* W = (c < 8) ? Wq : Wk;
        const float* a = attn + h * 2 * HD + ((c < 8) ? 0 : HD);
        float s = 0.f;
#pragma unroll
        for (int j = 0; j < HD; ++j) s += W[d * DIM + h * HD + j] * a[j];
        weff[i] = s;
    }
    if (t < 16) {
        int h = t & 7;
        const float* bb = (t < 8) ? bq : bk;
        const float* a  = attn + h * 2 * HD + ((t < 8) ? 0 : HD);
        float s = 0.f;
#pragma unroll
        for (int j = 0; j < HD; ++j) s += bb[h * HD + j] * a[j];
        beff[t] = s;
    }
}

// ---------- aq||ak GEMM: (80000 x 128) @ (128 x 16), one wave per 16-row tile ----------
__global__ void aqak_gemm(const float* __restrict__ X, const float* __restrict__ weff,
                          const float* __restrict__ beff, float* __restrict__ aqak) {
    __shared__ float sA[16 * LDSS];
    int row0 = blockIdx.x * 16;
    int tid  = threadIdx.x;     // 0..31
    for (int i = tid; i < 16 * 32; i += 32) {
        int r = i >> 5, c4 = i & 31;
        float4 v = *reinterpret_cast<const float4*>(X + (size_t)(row0 + r) * DIM + c4 * 4);
        float* d = sA + r * LDSS + c4 * 4;
        d[0] = v.x; d[1] = v.y; d[2] = v.z; d[3] = v.w;
    }
    __syncthreads();
    v8f acc = {};
#pragma unroll
    for (int k0 = 0; k0 < DIM; k0 += 4) {
        v2f a = ldsA(sA, tid, k0);
        v2f b = glbB(weff, 16, 0, tid, k0);
        acc = __builtin_amdgcn_wmma_f32_16x16x4_f32(false, a, false, b, (short)0, acc, false, false);
    }
    int c  = tid & 15;
    int r0 = row0 + ((tid >> 4) << 3);
    float bias = beff[c];
#pragma unroll
    for (int v = 0; v < 8; ++v) aqak[(size_t)(r0 + v) * 16 + c] = acc[v] + bias;
}

// ---------- V GEMM: (80000 x 128) @ (128 x 128), 8 waves per 16-row block ----------
__global__ void v_gemm(const float* __restrict__ X, const float* __restrict__ Wv,
                       const float* __restrict__ bv, float* __restrict__ V) {
    __shared__ float sA[16 * LDSS];
    int row0 = blockIdx.x * 16;
    int tid  = threadIdx.x;     // 0..255
    for (int i = tid; i < 16 * 32; i += 256) {
        int r = i >> 5, c4 = i & 31;
        float4 v = *reinterpret_cast<const float4*>(X + (size_t)(row0 + r) * DIM + c4 * 4);
        float* d = sA + r * LDSS + c4 * 4;
        d[0] = v.x; d[1] = v.y; d[2] = v.z; d[3] = v.w;
    }
    __syncthreads();
    int lane = tid & 31;
    int col0 = (tid >> 5) * 16;
    v8f acc = {};
#pragma unroll
    for (int k0 = 0; k0 < DIM; k0 += 4) {
        v2f a = ldsA(sA, lane, k0);
        v2f b = glbB(Wv, DIM, col0, lane, k0);
        acc = __builtin_amdgcn_wmma_f32_16x16x4_f32(false, a, false, b, (short)0, acc, false, false);
    }
    int c  = col0 + (lane & 15);
    int r0 = row0 + ((lane >> 4) << 3);
    float bias = bv[c];
#pragma unroll
    for (int v = 0; v < 8; ++v) V[(size_t)(r0 + v) * DIM + c] = acc[v] + bias;
}

// ---------- edge pass 1: per-target-node max (ordered-uint atomic max) ----------
__global__ void edge_max(const long long* __restrict__ eidx, const float* __restrict__ aqak,
                         unsigned int* __restrict__ m) {
    int t = blockIdx.x * blockDim.x + threadIdx.x;
    if (t >= NB * NE) return;
    int b = t / NE, e = t - b * NE;
    const long long* base = eidx + (size_t)b * 2 * NE;
    int sg = b * NN + (int)base[e];        // source node (global)
    int tg = b * NN + (int)base[NE + e];   // target node (global)
#pragma unroll
    for (int h = 0; h < NH; ++h) {
        float s = aqak[(size_t)tg * 16 + h] + aqak[(size_t)sg * 16 + 8 + h];
        s = (s > 0.f) ? s : 0.2f * s;      // leaky_relu(0.2)
        atomicMax(&m[tg * NH + h], fflip(s));
    }
}

// ---------- edge pass 2: exp-sum (den) + unnormalized weighted aggregation (into d_out) ----------
__global__ void edge_accum(const long long* __restrict__ eidx, const float* __restrict__ aqak,
                           const float* __restrict__ V, const unsigned int* __restrict__ m,
                           float* __restrict__ den, float* __restrict__ agg) {
    int t = blockIdx.x * blockDim.x + threadIdx.x;
    if (t >= NB * NE * NH) return;
    int h  = t & 7;
    int te = t >> 3;
    int b = te / NE, e = te - b * NE;
    const long long* base = eidx + (size_t)b * 2 * NE;
    int sg = b * NN + (int)base[e];
    int tg = b * NN + (int)base[NE + e];
    float s = aqak[(size_t)tg * 16 + h] + aqak[(size_t)sg * 16 + 8 + h];
    s = (s > 0.f) ? s : 0.2f * s;
    float ex = __expf(s - funflip(m[tg * NH + h]));
    atomicAdd(&den[tg * NH + h], ex);
    const float* vrow = V + (size_t)sg * DIM + h * HD;
    float*       arow = agg + (size_t)tg * DIM + h * HD;
#pragma unroll
    for (int d = 0; d < HD; ++d) atomicAdd(&arow[d], ex * vrow[d]);
}

// ---------- output: normalize agg by den, GEMM with Wo, +bo +residual, LayerNorm ----------
__global__ void out_gemm_ln(const float* __restrict__ X, const float* __restrict__ Wo,
                            const float* __restrict__ bo, const float* __restrict__ den,
                            const float* __restrict__ gamma, const float* __restrict__ beta,
                            float* __restrict__ out) {   // out holds agg on entry
    __shared__ float sT[16 * LDSS];
    int row0 = blockIdx.x * 16;
    int tid  = threadIdx.x;
    // stage normalized agg (per-head divide; empty segments: den==0 -> 0)
    for (int i = tid; i < 16 * 32; i += 256) {
        int r = i >> 5, c4 = i & 31;
        int rg = row0 + r;
        float4 v = *reinterpret_cast<const float4*>(out + (size_t)rg * DIM + c4 * 4);
        float dn = den[rg * NH + (c4 >> 2)];
        float inv = (dn > 0.f) ? 1.f / dn : 0.f;
        float* d = sT + r * LDSS + c4 * 4;
        d[0] = v.x * inv; d[1] = v.y * inv; d[2] = v.z * inv; d[3] = v.w * inv;
    }
    __syncthreads();
    int lane = tid & 31;
    int col0 = (tid >> 5) * 16;
    v8f acc = {};
#pragma unroll
    for (int k0 = 0; k0 < DIM; k0 += 4) {
        v2f a = ldsA(sT, lane, k0);
        v2f b = glbB(Wo, DIM, col0, lane, k0);
        acc = __builtin_amdgcn_wmma_f32_16x16x4_f32(false, a, false, b, (short)0, acc, false, false);
    }
    __syncthreads();   // everyone done reading A-tile
    int c  = col0 + (lane & 15);
    int r0 = (lane >> 4) << 3;
    float bias = bo[c];
#pragma unroll
    for (int v = 0; v < 8; ++v) {
        int r = r0 + v;
        sT[r * LDSS + c] = acc[v] + bias + X[(size_t)(row0 + r) * DIM + c];  // pre-LN
    }
    __syncthreads();
    // LayerNorm: wave w handles rows 2w, 2w+1; 4 cols per lane
    int w = tid >> 5;
#pragma unroll
    for (int rr = 0; rr < 2; ++rr) {
        int r = 2 * w + rr;
        float x0 = sT[r * LDSS + lane * 4 + 0];
        float x1 = sT[r * LDSS + lane * 4 + 1];
        float x2 = sT[r * LDSS + lane * 4 + 2];
        float x3 = sT[r * LDSS + lane * 4 + 3];
        float s = x0 + x1 + x2 + x3;
        float q = x0 * x0 + x1 * x1 + x2 * x2 + x3 * x3;
#pragma unroll
        for (int off = 16; off > 0; off >>= 1) {
            s += __shfl_xor(s, off, 32);
            q += __shfl_xor(q, off, 32);
        }
        float mu  = s * (1.f / DIM);
        float var = q * (1.f / DIM) - mu * mu;
        float rs  = rsqrtf(var + 1e-5f);
        size_t base = (size_t)(row0 + r) * DIM + lane * 4;
#pragma unroll
        for (int j = 0; j < 4; ++j) {
            int cc = lane * 4 + j;
            float xv = sT[r * LDSS + cc];
            out[base + j] = (xv - mu) * rs * gamma[cc] + beta[cc];
        }
    }
}

extern "C" void kernel_launch(void* const* d_in, const int* in_sizes, int n_in,
                              void* d_out, int out_size, void* d_ws, size_t ws_size,
                              hipStream_t stream) {
    const float*     X    = (const float*)d_in[0];
    const long long* eidx = (const long long*)d_in[1];
    const float* Wq   = (const float*)d_in[2];
    const float* bq   = (const float*)d_in[3];
    const float* Wk   = (const float*)d_in[4];
    const float* bk   = (const float*)d_in[5];
    const float* Wv   = (const float*)d_in[6];
    const float* bv   = (const float*)d_in[7];
    const float* attn = (const float*)d_in[8];
    const float* Wo   = (const float*)d_in[9];
    const float* bo   = (const float*)d_in[10];
    const float* gam  = (const float*)d_in[11];
    const float* bet  = (const float*)d_in[12];
    float* out = (float*)d_out;

    // workspace layout (floats): weff(2048) beff(16, pad to 2304) aqak(80000*16) V(80000*128) m den
    float* ws   = (float*)d_ws;
    float* weff = ws;
    float* beff = ws + 2048;
    float* aqak = ws + 2304;
    float* V    = aqak + (size_t)NB * NN * 16;
    unsigned int* m = (unsigned int*)(V + (size_t)NB * NN * DIM);
    float* den  = (float*)(m + (size_t)NB * NN * NH);

    hipMemsetAsync(out, 0, (size_t)out_size * sizeof(float), stream);       // agg accumulator
    hipMemsetAsync(m,   0, (size_t)NB * NN * NH * sizeof(unsigned int), stream); // enc(-inf-ish)
    hipMemsetAsync(den, 0, (size_t)NB * NN * NH * sizeof(float), stream);

    prep_weff<<<1, 256, 0, stream>>>(Wq, bq, Wk, bk, attn, weff, beff);
    aqak_gemm<<<(NB * NN) / 16, 32, 0, stream>>>(X, weff, beff, aqak);
    v_gemm<<<(NB * NN) / 16, 256, 0, stream>>>(X, Wv, bv, V);
    edge_max<<<(NB * NE + 255) / 256, 256, 0, stream>>>(eidx, aqak, m);
    edge_accum<<<(NB * NE * NH + 255) / 256, 256, 0, stream>>>(eidx, aqak, V, m, den, out);
    out_gemm_ln<<<(NB * NN) / 16, 256, 0, stream>>>(X, Wo, bo, den, gam, bet, out);
}